// Transformer_37795712204871
// MI455X (gfx1250) — compile-verified
//
#include <hip/hip_runtime.h>

// ---------------------------------------------------------------------------
// Transformer forward (fp32) for gfx1250 using V_WMMA_F32_16X16X4_F32.
// GEMM: 32x64 tile per wave (8 accumulators), software-pipelined K loop
// (register double-buffering) so loads for step k+4 are in flight while the
// matrix pipe executes step k.
// ---------------------------------------------------------------------------

typedef __attribute__((ext_vector_type(2))) float v2f;
typedef __attribute__((ext_vector_type(8))) float v8f;

#define DIMSZ  512
#define SEQ    512
#define NB     4
#define NH     8
#define HD     64
#define NLAYER 6
#define ROWS   (NB * SEQ)   // 2048
#define VOCABN 32000

__device__ __forceinline__ v8f wmma4(v2f a, v2f b, v8f c) {
  // D = A(16x4,f32) * B(4x16,f32) + C(16x16,f32)
  return __builtin_amdgcn_wmma_f32_16x16x4_f32(false, a, false, b, (short)0, c,
                                               false, false);
}

// ---------------------------------------------------------------------------
// Flexible fp32 WMMA GEMM.
//   C = act( alpha * (A@B [+Cadd]) + bias )
// Per wave: 32x64 tile of C (2 A-frags x 4 B-frags = 8 WMMAs / k-step).
// Block: 128 threads = 4 waves stacked in M -> 128x64 block tile.
// transB: B is stored N x K row-major (used for Q.K^T and x.emb^T).
// cmode 0: C[z*sC + m*ldc + n]
// cmode 1: projection scatter to (B,H,L,D):  m=(b*512+l), n=(h*64+d)
// cmode 2: attn-out scatter to concat (b,q, d*H+h):  z=(b*8+h), m=q, n=d
// ---------------------------------------------------------------------------
__global__ __launch_bounds__(128) void gemm_f32_wmma(
    const float* __restrict__ A, const float* __restrict__ Bm,
    const float* __restrict__ bias, const float* __restrict__ Cadd,
    float* __restrict__ C, int M, int N, int K, int lda, int ldb, int ldc,
    int ldcadd, long long sA, long long sB, long long sC, long long sCadd,
    int transB, int cmode, int relu, int caddModH, float alpha) {
  const int z    = blockIdx.z;
  const int wave = threadIdx.x >> 5;
  const int lane = threadIdx.x & 31;
  const int half = lane >> 4;   // K-pair select (K 0..1 vs 2..3)
  const int lr   = lane & 15;   // row (A) / col (B,C) within tile

  const int mBase = blockIdx.y * 128 + wave * 32;
  const int nBase = blockIdx.x * 64;

  const float* Ab = A + (long long)z * sA;
  const float* Bb = Bm + (long long)z * sB;

  const v8f zero8 = {0.f, 0.f, 0.f, 0.f, 0.f, 0.f, 0.f, 0.f};
  v8f acc[8];  // [mSub*4 + nSub]
#pragma unroll
  for (int i = 0; i < 8; ++i) acc[i] = zero8;

  // A fragment rows for the two 16-row sub-tiles of this wave.
  const float* aR0 = Ab + (long long)(mBase + lr) * lda + 2 * half;
  const float* aR1 = aR0 + (long long)16 * lda;

  if (transB) {
    const float* b0p = Bb + (long long)(nBase + 0 + lr) * ldb + 2 * half;
    const float* b1p = Bb + (long long)(nBase + 16 + lr) * ldb + 2 * half;
    const float* b2p = Bb + (long long)(nBase + 32 + lr) * ldb + 2 * half;
    const float* b3p = Bb + (long long)(nBase + 48 + lr) * ldb + 2 * half;
    // Prologue: fragments for k=0.
    v2f a0 = *(const v2f*)(aR0);
    v2f a1 = *(const v2f*)(aR1);
    v2f b0 = *(const v2f*)(b0p);
    v2f b1 = *(const v2f*)(b1p);
    v2f b2 = *(const v2f*)(b2p);
    v2f b3 = *(const v2f*)(b3p);
    for (int kk = 0; kk + 4 < K; kk += 4) {
      const int kn = kk + 4;
      // Prefetch next k-step fragments (independent of current WMMAs).
      v2f a0n = *(const v2f*)(aR0 + kn);
      v2f a1n = *(const v2f*)(aR1 + kn);
      v2f b0n = *(const v2f*)(b0p + kn);
      v2f b1n = *(const v2f*)(b1p + kn);
      v2f b2n = *(const v2f*)(b2p + kn);
      v2f b3n = *(const v2f*)(b3p + kn);
      acc[0] = wmma4(a0, b0, acc[0]);
      acc[1] = wmma4(a0, b1, acc[1]);
      acc[2] = wmma4(a0, b2, acc[2]);
      acc[3] = wmma4(a0, b3, acc[3]);
      acc[4] = wmma4(a1, b0, acc[4]);
      acc[5] = wmma4(a1, b1, acc[5]);
      acc[6] = wmma4(a1, b2, acc[6]);
      acc[7] = wmma4(a1, b3, acc[7]);
      a0 = a0n; a1 = a1n; b0 = b0n; b1 = b1n; b2 = b2n; b3 = b3n;
    }
    acc[0] = wmma4(a0, b0, acc[0]);
    acc[1] = wmma4(a0, b1, acc[1]);
    acc[2] = wmma4(a0, b2, acc[2]);
    acc[3] = wmma4(a0, b3, acc[3]);
    acc[4] = wmma4(a1, b0, acc[4]);
    acc[5] = wmma4(a1, b1, acc[5]);
    acc[6] = wmma4(a1, b2, acc[6]);
    acc[7] = wmma4(a1, b3, acc[7]);
  } else {
    const float* bCol = Bb + nBase + lr;
    // Prologue fragments for k=0: B rows (2*half) and (2*half+1).
    const float* r0 = bCol + (long long)(2 * half) * ldb;
    const float* r1 = r0 + ldb;
    v2f a0 = *(const v2f*)(aR0);
    v2f a1 = *(const v2f*)(aR1);
    v2f b0; b0.x = r0[0];  b0.y = r1[0];
    v2f b1; b1.x = r0[16]; b1.y = r1[16];
    v2f b2; b2.x = r0[32]; b2.y = r1[32];
    v2f b3; b3.x = r0[48]; b3.y = r1[48];
    for (int kk = 0; kk + 4 < K; kk += 4) {
      const int kn = kk + 4;
      const float* n0 = bCol + (long long)(kn + 2 * half) * ldb;
      const float* n1 = n0 + ldb;
      v2f a0n = *(const v2f*)(aR0 + kn);
      v2f a1n = *(const v2f*)(aR1 + kn);
      v2f b0n; b0n.x = n0[0];  b0n.y = n1[0];
      v2f b1n; b1n.x = n0[16]; b1n.y = n1[16];
      v2f b2n; b2n.x = n0[32]; b2n.y = n1[32];
      v2f b3n; b3n.x = n0[48]; b3n.y = n1[48];
      acc[0] = wmma4(a0, b0, acc[0]);
      acc[1] = wmma4(a0, b1, acc[1]);
      acc[2] = wmma4(a0, b2, acc[2]);
      acc[3] = wmma4(a0, b3, acc[3]);
      acc[4] = wmma4(a1, b0, acc[4]);
      acc[5] = wmma4(a1, b1, acc[5]);
      acc[6] = wmma4(a1, b2, acc[6]);
      acc[7] = wmma4(a1, b3, acc[7]);
      a0 = a0n; a1 = a1n; b0 = b0n; b1 = b1n; b2 = b2n; b3 = b3n;
    }
    acc[0] = wmma4(a0, b0, acc[0]);
    acc[1] = wmma4(a0, b1, acc[1]);
    acc[2] = wmma4(a0, b2, acc[2]);
    acc[3] = wmma4(a0, b3, acc[3]);
    acc[4] = wmma4(a1, b0, acc[4]);
    acc[5] = wmma4(a1, b1, acc[5]);
    acc[6] = wmma4(a1, b2, acc[6]);
    acc[7] = wmma4(a1, b3, acc[7]);
  }

  const bool useCadd = (Cadd != nullptr);
  long long caddOff = 0;
  if (useCadd) caddOff = (long long)(caddModH ? (z & 7) : z) * sCadd;

#pragma unroll
  for (int ms = 0; ms < 2; ++ms) {
    const int mSub = mBase + ms * 16;
#pragma unroll
    for (int j = 0; j < 4; ++j) {
      const int ncol = nBase + j * 16 + lr;
      const float bb = bias ? bias[ncol] : 0.0f;
      const v8f a = acc[ms * 4 + j];
#pragma unroll
      for (int r = 0; r < 8; ++r) {
        const int mrow = mSub + r + 8 * half;
        float v = a[r];
        if (useCadd) v += Cadd[caddOff + (long long)mrow * ldcadd + ncol];
        v = alpha * v + bb;
        if (relu && v < 0.0f) v = 0.0f;
        long long idx;
        if (cmode == 0) {
          idx = (long long)z * sC + (long long)mrow * ldc + ncol;
        } else if (cmode == 1) {
          const int bI = mrow >> 9, l = mrow & 511;
          const int h = ncol >> 6, d = ncol & 63;
          idx = ((long long)((bI * NH + h) * SEQ + l)) * HD + d;
        } else {
          const int bI = z >> 3, h = z & 7;
          idx = ((long long)(bI * SEQ + mrow)) * DIMSZ + ncol * NH + h;
        }
        C[idx] = v;
      }
    }
  }
}

// ---------------------------------------------------------------------------
// Masked row softmax over (B,H,L,L); one wave32 per 512-wide row.
// keytok masks key positions (token != PAD); causal adds k<=q.
// ---------------------------------------------------------------------------
__global__ __launch_bounds__(128) void softmax_kernel(
    float* __restrict__ S, const int* __restrict__ keytok, int causal) {
  const int row  = blockIdx.x * 4 + (threadIdx.x >> 5);  // (b*8+h)*512+q
  const int lane = threadIdx.x & 31;
  const int q = row & 511;
  const int b = row >> 12;  // / (8*512)
  float* p = S + (long long)row * SEQ;
  const int* tb = keytok + b * SEQ;

  float v[16];
  float mx = -3.4e38f;
#pragma unroll
  for (int i = 0; i < 16; ++i) {
    const int k = lane + 32 * i;
    const bool valid = (tb[k] != 0) && (!causal || k <= q);
    const float x = valid ? p[k] : -3.4e38f;
    v[i] = x;
    mx = fmaxf(mx, x);
  }
  for (int m = 16; m; m >>= 1) mx = fmaxf(mx, __shfl_xor(mx, m, 32));
  float sum = 0.f;
#pragma unroll
  for (int i = 0; i < 16; ++i) {
    const float e = (v[i] > -1.0e37f) ? __expf(v[i] - mx) : 0.0f;
    v[i] = e;
    sum += e;
  }
  for (int m = 16; m; m >>= 1) sum += __shfl_xor(sum, m, 32);
  const float inv = 1.0f / sum;
#pragma unroll
  for (int i = 0; i < 16; ++i) p[lane + 32 * i] = v[i] * inv;
}

// ---------------------------------------------------------------------------
// y = LayerNorm(x + resid) * g + b ; one wave32 per 512-wide row.
// ---------------------------------------------------------------------------
__global__ __launch_bounds__(128) void ln_kernel(
    const float* __restrict__ X, const float* __restrict__ R,
    const float* __restrict__ g, const float* __restrict__ bta,
    float* __restrict__ Y) {
  const int row  = blockIdx.x * 4 + (threadIdx.x >> 5);
  const int lane = threadIdx.x & 31;
  const float* xr = X + (long long)row * DIMSZ;
  const float* rr = R + (long long)row * DIMSZ;
  float v[16];
  float s = 0.f;
#pragma unroll
  for (int i = 0; i < 16; ++i) {
    v[i] = xr[lane + 32 * i] + rr[lane + 32 * i];
    s += v[i];
  }
  for (int m = 16; m; m >>= 1) s += __shfl_xor(s, m, 32);
  const float mean = s * (1.0f / DIMSZ);
  float var = 0.f;
#pragma unroll
  for (int i = 0; i < 16; ++i) {
    const float d = v[i] - mean;
    var += d * d;
  }
  for (int m = 16; m; m >>= 1) var += __shfl_xor(var, m, 32);
  var *= (1.0f / DIMSZ);
  const float rstd = rsqrtf(var + 1e-5f);
  float* yr = Y + (long long)row * DIMSZ;
#pragma unroll
  for (int i = 0; i < 16; ++i) {
    const int k = lane + 32 * i;
    yr[k] = (v[i] - mean) * rstd * g[k] + bta[k];
  }
}

// x[b,l,:] = emb[tok[b,l],:] * sqrt(DIM)
__global__ void embed_kernel(const int* __restrict__ tok,
                             const float* __restrict__ emb,
                             float* __restrict__ out) {
  const int i = blockIdx.x * blockDim.x + threadIdx.x;  // ROWS*DIMSZ total
  const int t = tok[i >> 9];
  out[i] = emb[(long long)t * DIMSZ + (i & 511)] * 22.62741699796952f;
}

// Sinusoidal PE table (SEQ x DIM)
__global__ void pe_kernel(float* __restrict__ pe) {
  const int i = blockIdx.x * blockDim.x + threadIdx.x;  // SEQ*DIMSZ total
  const int p = i >> 9, d = i & 511;
  const float e = (float)(d >> 1) / (float)DIMSZ;
  const float freq = __powf(10000.0f, e);
  const float a = (float)p / freq;
  pe[i] = (d & 1) ? cosf(a) : sinf(a);
}

// ---------------------------------------------------------------------------
// Host orchestration
// ---------------------------------------------------------------------------
extern "C" void kernel_launch(void* const* d_in, const int* in_sizes, int n_in,
                              void* d_out, int out_size, void* d_ws,
                              size_t ws_size, hipStream_t stream) {
  const int*   tok_in  = (const int*)d_in[0];
  const int*   tok_out = (const int*)d_in[1];
  const float* emb     = (const float*)d_in[2];
  const float* fc_b    = (const float*)d_in[3];

  // Stacked param dicts flatten in sorted-key order (jax.tree.map rebuilds
  // dicts with sorted keys). enc_params leaves:
  enum { E_f1W, E_f1b, E_f2W, E_f2b, E_fln_b, E_fln_g, E_kW, E_kb, E_ln_b,
         E_ln_g, E_oW, E_ob, E_pkW, E_pkb, E_pqW, E_pqb, E_qW, E_qb, E_vW,
         E_vb, E_COUNT };
  // dec_params leaves:
  enum { D_c_kW, D_c_kb, D_c_ln_b, D_c_ln_g, D_c_oW, D_c_ob, D_c_pkW,
         D_c_pkb, D_c_pqW, D_c_pqb, D_c_qW, D_c_qb, D_c_vW, D_c_vb, D_f1W,
         D_f1b, D_f2W, D_f2b, D_fln_b, D_fln_g, D_s_kW, D_s_kb, D_s_ln_b,
         D_s_ln_g, D_s_oW, D_s_ob, D_s_pkW, D_s_pkb, D_s_pqW, D_s_pqb,
         D_s_qW, D_s_qb, D_s_vW, D_s_vb, D_COUNT };

  auto encP = [&](int leaf, int layer) -> const float* {
    const int idx = 4 + leaf;
    return (const float*)d_in[idx] +
           (size_t)layer * ((size_t)in_sizes[idx] / NLAYER);
  };
  auto decP = [&](int leaf, int layer) -> const float* {
    const int idx = 4 + E_COUNT + leaf;
    return (const float*)d_in[idx] +
           (size_t)layer * ((size_t)in_sizes[idx] / NLAYER);
  };

  // ---- workspace layout (floats) ----
  float* w = (float*)d_ws;
  float* pe     = w; w += (size_t)SEQ * DIMSZ;          // 262144
  float* enc_x  = w; w += (size_t)ROWS * DIMSZ;         // 1M
  float* dec_x  = w; w += (size_t)ROWS * DIMSZ;
  float* Qb     = w; w += (size_t)ROWS * DIMSZ;         // (B,H,L,D)
  float* Kb     = w; w += (size_t)ROWS * DIMSZ;
  float* Vb     = w; w += (size_t)ROWS * DIMSZ;
  float* PQ     = w; w += (size_t)NH * SEQ * HD;        // (H,L,D)
  float* PK     = w; w += (size_t)NH * SEQ * HD;
  float* pbias  = w; w += (size_t)NH * SEQ * SEQ;       // PQ.PK^T per head
  float* scores = w; w += (size_t)NB * NH * SEQ * SEQ;  // 32MB
  float* concat = w; w += (size_t)ROWS * DIMSZ;
  float* tmp1   = w; w += (size_t)ROWS * DIMSZ;
  float* ffh    = w; w += (size_t)ROWS * HD;

  auto gemm = [&](const float* A, const float* B, const float* bias,
                  const float* cadd, float* C, int M, int N, int K, int lda,
                  int ldb, int ldc, int ldcadd, long long sA, long long sB,
                  long long sC, long long sCadd, int batches, int transB,
                  int cmode, int relu, int caddModH, float alpha) {
    dim3 g(N / 64, M / 128, batches), blk(128, 1, 1);
    gemm_f32_wmma<<<g, blk, 0, stream>>>(A, B, bias, cadd, C, M, N, K, lda,
                                         ldb, ldc, ldcadd, sA, sB, sC, sCadd,
                                         transB, cmode, relu, caddModH, alpha);
  };

  auto mha = [&](float* xq, const float* xkv, const int* keytok, int causal,
                 const float* qW, const float* qb, const float* kW,
                 const float* kb, const float* vW, const float* vb,
                 const float* pqW, const float* pqb, const float* pkW,
                 const float* pkb, const float* oW, const float* ob,
                 const float* lng, const float* lnb) {
    // PE projections -> (H,L,D)
    gemm(pe, pqW, pqb, nullptr, PQ, SEQ, DIMSZ, DIMSZ, DIMSZ, DIMSZ, 0, 0,
         0, 0, 0, 0, 1, 0, 1, 0, 0, 1.0f);
    gemm(pe, pkW, pkb, nullptr, PK, SEQ, DIMSZ, DIMSZ, DIMSZ, DIMSZ, 0, 0,
         0, 0, 0, 0, 1, 0, 1, 0, 0, 1.0f);
    // pbias[h] = PQ[h] @ PK[h]^T   (batched over H)
    gemm(PQ, PK, nullptr, nullptr, pbias, SEQ, SEQ, HD, HD, HD, SEQ, 0,
         (long long)SEQ * HD, (long long)SEQ * HD, (long long)SEQ * SEQ, 0,
         NH, 1, 0, 0, 0, 1.0f);
    // Q,K,V projections -> (B,H,L,D)
    gemm(xq, qW, qb, nullptr, Qb, ROWS, DIMSZ, DIMSZ, DIMSZ, DIMSZ, 0, 0,
         0, 0, 0, 0, 1, 0, 1, 0, 0, 1.0f);
    gemm(xkv, kW, kb, nullptr, Kb, ROWS, DIMSZ, DIMSZ, DIMSZ, DIMSZ, 0, 0,
         0, 0, 0, 0, 1, 0, 1, 0, 0, 1.0f);
    gemm(xkv, vW, vb, nullptr, Vb, ROWS, DIMSZ, DIMSZ, DIMSZ, DIMSZ, 0, 0,
         0, 0, 0, 0, 1, 0, 1, 0, 0, 1.0f);
    // scores = (Q.K^T + pbias[h]) * ADIM^-0.5   (batched over B*H)
    gemm(Qb, Kb, nullptr, pbias, scores, SEQ, SEQ, HD, HD, HD, SEQ, SEQ,
         (long long)SEQ * HD, (long long)SEQ * HD, (long long)SEQ * SEQ,
         (long long)SEQ * SEQ, NB * NH, 1, 0, 0, 1, 0.125f);
    softmax_kernel<<<dim3(NB * NH * SEQ / 4), dim3(128), 0, stream>>>(
        scores, keytok, causal);
    // attn @ V -> concat in (b, q, d*H+h) ordering
    gemm(scores, Vb, nullptr, nullptr, concat, SEQ, HD, SEQ, SEQ, HD, 0, 0,
         (long long)SEQ * SEQ, (long long)SEQ * HD, 0, 0, NB * NH, 0, 2, 0, 0,
         1.0f);
    // output projection
    gemm(concat, oW, ob, nullptr, tmp1, ROWS, DIMSZ, DIMSZ, DIMSZ, DIMSZ,
         DIMSZ, 0, 0, 0, 0, 0, 1, 0, 0, 0, 0, 1.0f);
    // residual + layernorm (in place on xq)
    ln_kernel<<<dim3(ROWS / 4), dim3(128), 0, stream>>>(tmp1, xq, lng, lnb,
                                                        xq);
  };

  auto ff = [&](float* x, const float* f1W, const float* f1b,
                const float* f2W, const float* f2b, const float* g,
                const float* b) {
    gemm(x, f1W, f1b, nullptr, ffh, ROWS, HD, DIMSZ, DIMSZ, HD, HD, 0, 0, 0,
         0, 0, 1, 0, 0, 1, 0, 1.0f);  // relu
    gemm(ffh, f2W, f2b, nullptr, tmp1, ROWS, DIMSZ, HD, HD, DIMSZ, DIMSZ, 0,
         0, 0, 0, 0, 1, 0, 0, 0, 0, 1.0f);
    ln_kernel<<<dim3(ROWS / 4), dim3(128), 0, stream>>>(tmp1, x, g, b, x);
  };

  // ---- setup ----
  pe_kernel<<<dim3(SEQ * DIMSZ / 256), dim3(256), 0, stream>>>(pe);
  embed_kernel<<<dim3(ROWS * DIMSZ / 256), dim3(256), 0, stream>>>(
      tok_in, emb, enc_x);
  embed_kernel<<<dim3(ROWS * DIMSZ / 256), dim3(256), 0, stream>>>(
      tok_out, emb, dec_x);

  // ---- encoder ----
  for (int i = 0; i < NLAYER; ++i) {
    mha(enc_x, enc_x, tok_in, 0, encP(E_qW, i), encP(E_qb, i), encP(E_kW, i),
        encP(E_kb, i), encP(E_vW, i), encP(E_vb, i), encP(E_pqW, i),
        encP(E_pqb, i), encP(E_pkW, i), encP(E_pkb, i), encP(E_oW, i),
        encP(E_ob, i), encP(E_ln_g, i), encP(E_ln_b, i));
    ff(enc_x, encP(E_f1W, i), encP(E_f1b, i), encP(E_f2W, i), encP(E_f2b, i),
       encP(E_fln_g, i), encP(E_fln_b, i));
  }

  // ---- decoder ----
  for (int i = 0; i < NLAYER; ++i) {
    mha(dec_x, dec_x, tok_out, 1, decP(D_s_qW, i), decP(D_s_qb, i),
        decP(D_s_kW, i), decP(D_s_kb, i), decP(D_s_vW, i), decP(D_s_vb, i),
        decP(D_s_pqW, i), decP(D_s_pqb, i), decP(D_s_pkW, i),
        decP(D_s_pkb, i), decP(D_s_oW, i), decP(D_s_ob, i),
        decP(D_s_ln_g, i), decP(D_s_ln_b, i));
    mha(dec_x, enc_x, tok_in, 0, decP(D_c_qW, i), decP(D_c_qb, i),
        decP(D_c_kW, i), decP(D_c_kb, i), decP(D_c_vW, i), decP(D_c_vb, i),
        decP(D_c_pqW, i), decP(D_c_pqb, i), decP(D_c_pkW, i),
        decP(D_c_pkb, i), decP(D_c_oW, i), decP(D_c_ob, i),
        decP(D_c_ln_g, i), decP(D_c_ln_b, i));
    ff(dec_x, decP(D_f1W, i), decP(D_f1b, i), decP(D_f2W, i), decP(D_f2b, i),
       decP(D_fln_g, i), decP(D_fln_b, i));
  }

  // ---- logits = dec_x @ emb^T + fc_b ----
  gemm(dec_x, emb, fc_b, nullptr, (float*)d_out, ROWS, VOCABN, DIMSZ, DIMSZ,
       DIMSZ, VOCABN, 0, 0, 0, 0, 0, 1, 1, 0, 0, 0, 1.0f);
}